// EmbeddingLoss_4896262717860
// MI455X (gfx1250) — compile-verified
//
#include <hip/hip_runtime.h>
#include <math.h>

typedef __attribute__((ext_vector_type(2))) float v2f;
typedef __attribute__((ext_vector_type(8))) float v8f;

#define NFDIM 64
#define NCLUST 128
#define DVAR 0.5f
#define DDIST 3.0f

// ws layout (floats):
//   [0, 128)           counts
//   [128, 128+8192)    sums (cluster-major, 128 x 64)
//   [8320, 8448)       varsum
#define WS_COUNTS 0
#define WS_SUMS   NCLUST
#define WS_VAR    (NCLUST + NCLUST * NFDIM)
#define WS_TOTAL  (WS_VAR + NCLUST)

__global__ void k_zero(float* __restrict__ ws) {
  int i = blockIdx.x * blockDim.x + threadIdx.x;
  if (i < WS_TOTAL) ws[i] = 0.0f;
}

// Pass 1: per-cluster counts and feature sums. LDS-first (stride 65 to spread
// banks: addr = id*65+k -> bank (id+k)&63), then global atomic flush.
__global__ void __launch_bounds__(256) k_pass1(const float4* __restrict__ feats4,
                                               const int* __restrict__ labels,
                                               float* __restrict__ ws, int N) {
  __shared__ float lsum[NCLUST * 65];
  __shared__ float lcnt[NCLUST];
  const int tid = threadIdx.x;
  for (int i = tid; i < NCLUST * 65; i += blockDim.x) lsum[i] = 0.0f;
  for (int i = tid; i < NCLUST; i += blockDim.x) lcnt[i] = 0.0f;
  __syncthreads();

  const int stride = gridDim.x * blockDim.x;
  for (int p = blockIdx.x * blockDim.x + tid; p < N; p += stride) {
    const int id = labels[2 * p + 1];
    float* dst = &lsum[id * 65];
#pragma unroll
    for (int k = 0; k < 16; ++k) {
      float4 v = feats4[p * 16 + k];
      atomicAdd(&dst[4 * k + 0], v.x);
      atomicAdd(&dst[4 * k + 1], v.y);
      atomicAdd(&dst[4 * k + 2], v.z);
      atomicAdd(&dst[4 * k + 3], v.w);
    }
    atomicAdd(&lcnt[id], 1.0f);
  }
  __syncthreads();

  for (int i = tid; i < NCLUST * NFDIM; i += blockDim.x) {
    int c = i >> 6, k = i & 63;
    atomicAdd(&ws[WS_SUMS + i], lsum[c * 65 + k]);
  }
  for (int i = tid; i < NCLUST; i += blockDim.x)
    atomicAdd(&ws[WS_COUNTS + i], lcnt[i]);
}

// Pass 2: per-point relu(||x - mean_id|| - 0.5)^2, accumulated per cluster.
__global__ void __launch_bounds__(256) k_pass2(const float4* __restrict__ feats4,
                                               const int* __restrict__ labels,
                                               float* __restrict__ ws, int N) {
  __shared__ float lmean[NCLUST * 65];
  __shared__ float lvar[NCLUST];
  const int tid = threadIdx.x;
  for (int i = tid; i < NCLUST * NFDIM; i += blockDim.x) {
    int c = i >> 6, k = i & 63;
    lmean[c * 65 + k] = ws[WS_SUMS + i] / ws[WS_COUNTS + c];
  }
  for (int i = tid; i < NCLUST; i += blockDim.x) lvar[i] = 0.0f;
  __syncthreads();

  const int stride = gridDim.x * blockDim.x;
  for (int p = blockIdx.x * blockDim.x + tid; p < N; p += stride) {
    const int id = labels[2 * p + 1];
    const float* mu = &lmean[id * 65];
    float s = 0.0f;
#pragma unroll
    for (int k = 0; k < 16; ++k) {
      float4 v = feats4[p * 16 + k];
      float d0 = v.x - mu[4 * k + 0];
      float d1 = v.y - mu[4 * k + 1];
      float d2 = v.z - mu[4 * k + 2];
      float d3 = v.w - mu[4 * k + 3];
      s += d0 * d0 + d1 * d1 + d2 * d2 + d3 * d3;
    }
    float t = sqrtf(s) - DVAR;
    if (t > 0.0f) atomicAdd(&lvar[id], t * t);
  }
  __syncthreads();

  for (int i = tid; i < NCLUST; i += blockDim.x)
    atomicAdd(&ws[WS_VAR + i], lvar[i]);
}

// Final: Gram matrix of means via V_WMMA_F32_16X16X4_F32, distance loss,
// variance loss, scalar output. One block, 8 waves; wave w owns row-tile w.
__global__ void __launch_bounds__(256) k_final(const float* __restrict__ ws,
                                               float* __restrict__ out) {
  __shared__ float lm[NCLUST * 66];  // stride 66: float2-aligned, conflict-free
  __shared__ float n2[NCLUST];
  __shared__ float red[256];
  __shared__ float dist_saved;
  const int tid = threadIdx.x;

  for (int i = tid; i < NCLUST * NFDIM; i += 256) {
    int c = i >> 6, k = i & 63;
    lm[c * 66 + k] = ws[WS_SUMS + i] / ws[WS_COUNTS + c];
  }
  __syncthreads();
  if (tid < NCLUST) {
    float s = 0.0f;
    for (int k = 0; k < NFDIM; ++k) { float v = lm[tid * 66 + k]; s += v * v; }
    n2[tid] = s;
  }
  __syncthreads();

  const int wave = tid >> 5;
  const int lane = tid & 31;
  const int half = lane >> 4;   // K half-select for A/B operands
  const int mrow = lane & 15;   // row (A) / col (B) within tile
  const int ti = wave;          // 8 waves cover 8 row-tiles

  float acc = 0.0f;
  for (int tj = 0; tj < 8; ++tj) {
    v8f cacc = {};
#pragma unroll
    for (int k0 = 0; k0 < NFDIM; k0 += 4) {
      const int kk = k0 + half * 2;
      const float* ap = &lm[(ti * 16 + mrow) * 66 + kk];
      const float* bp = &lm[(tj * 16 + mrow) * 66 + kk];
      v2f a, b;
      a.x = ap[0]; a.y = ap[1];
      b.x = bp[0]; b.y = bp[1];
      // D = A(16x4) * B(4x16) + C, f32, full precision Gram tile
      cacc = __builtin_amdgcn_wmma_f32_16x16x4_f32(
          false, a, false, b, (short)0, cacc, false, false);
    }
#pragma unroll
    for (int r = 0; r < 8; ++r) {
      const int i = ti * 16 + r + half * 8;  // C/D layout: VGPR r -> M=r (lo), r+8 (hi)
      const int j = tj * 16 + mrow;          // N = lane % 16
      if (i != j) {
        float sq = n2[i] + n2[j] - 2.0f * cacc[r];
        sq = fmaxf(sq, 0.0f);
        float t = DDIST - sqrtf(sq);
        if (t > 0.0f) acc += t * t;
      }
    }
  }

  red[tid] = acc;
  __syncthreads();
  for (int s = 128; s > 0; s >>= 1) {
    if (tid < s) red[tid] += red[tid + s];
    __syncthreads();
  }
  if (tid == 0) dist_saved = red[0];
  __syncthreads();

  red[tid] = (tid < NCLUST) ? (ws[WS_VAR + tid] / ws[WS_COUNTS + tid]) : 0.0f;
  __syncthreads();
  for (int s = 128; s > 0; s >>= 1) {
    if (tid < s) red[tid] += red[tid + s];
    __syncthreads();
  }
  if (tid == 0) {
    out[0] = red[0] / (float)NCLUST +
             dist_saved / (float)(NCLUST * (NCLUST - 1));
  }
}

extern "C" void kernel_launch(void* const* d_in, const int* in_sizes, int n_in,
                              void* d_out, int out_size, void* d_ws, size_t ws_size,
                              hipStream_t stream) {
  const float* feats = (const float*)d_in[0];
  const int* labels  = (const int*)d_in[1];
  float* out = (float*)d_out;
  float* ws  = (float*)d_ws;
  const int N = in_sizes[0] / NFDIM;

  k_zero<<<(WS_TOTAL + 255) / 256, 256, 0, stream>>>(ws);

  const int blocks = 768;  // grid-stride; plenty of waves to saturate HBM
  k_pass1<<<blocks, 256, 0, stream>>>((const float4*)feats, labels, ws, N);
  k_pass2<<<blocks, 256, 0, stream>>>((const float4*)feats, labels, ws, N);
  k_final<<<1, 256, 0, stream>>>(ws, out);
}